// AirspaceGATv2_71064528879806
// MI455X (gfx1250) — compile-verified
//
#include <hip/hip_runtime.h>
#include <hip/hip_bf16.h>
#include <math.h>

// ---------------------------------------------------------------------------
// GATv2 x2 (PyG-style) forward for MI455X / gfx1250, wave32.
// GEMMs: V_WMMA_F32_16X16X4_F32 with 2x4 register tile blocking (8 WMMA /
// K-step, ~10.7 FLOP/B, ~110 VGPRs -> no spills). Edge softmax/aggregation:
// float4 gathers + L2-resident f32 atomics.
// ---------------------------------------------------------------------------

typedef float v2f __attribute__((ext_vector_type(2)));
typedef float v8f __attribute__((ext_vector_type(8)));

__device__ __forceinline__ float waveSum(float v) {
#pragma unroll
  for (int off = 16; off; off >>= 1) v += __shfl_xor(v, off, 32);
  return v;
}

__device__ __forceinline__ void atomicAddF(float* p, float v) {
  (void)__hip_atomic_fetch_add(p, v, __ATOMIC_RELAXED, __HIP_MEMORY_SCOPE_AGENT);
}

// float atomic max via signed/unsigned integer monotonicity trick
__device__ __forceinline__ void atomicMaxF(float* p, float v) {
  if (!(v < 0.0f)) (void)atomicMax((int*)p, __float_as_int(v));
  else             (void)atomicMin((unsigned int*)p, __float_as_uint(v));
}

// ---------------- utility fills ----------------
__global__ void k_fill(float* __restrict__ p, float val, long long n) {
  long long i = (long long)blockIdx.x * blockDim.x + threadIdx.x;
  long long s = (long long)gridDim.x * blockDim.x;
  for (; i < n; i += s) p[i] = val;
}

__global__ void k_bias_init(float* __restrict__ out, const float* __restrict__ bias,
                            long long n, int F) {
  long long i = (long long)blockIdx.x * blockDim.x + threadIdx.x;
  long long s = (long long)gridDim.x * blockDim.x;
  for (; i < n; i += s) out[i] = bias[i % F];
}

// ---------------- edge-attr column mean (ED = 4) ----------------
__global__ void k_ea_accum(const float* __restrict__ eattr, float* __restrict__ eaSum,
                           int E) {
  const int lane = threadIdx.x & 31;
  const float4* ea4 = (const float4*)eattr;
  float loc[4] = {0.f, 0.f, 0.f, 0.f};
  long long i = (long long)blockIdx.x * blockDim.x + threadIdx.x;
  long long s = (long long)gridDim.x * blockDim.x;
  for (; i < E; i += s) {
    float4 v = ea4[i];
    loc[0] += v.x; loc[1] += v.y; loc[2] += v.z; loc[3] += v.w;
  }
#pragma unroll
  for (int k = 0; k < 4; ++k) {
    float v = waveSum(loc[k]);
    if (lane == 0) atomicAddF(&eaSum[k], v);
  }
}

// ---------------- f32 WMMA GEMM: C[M,Nn] = A[M,K] @ W[K,Nn] ----------------
// One wave computes a 32x64 block (2x4 tiles of 16x16); K stepped by 4.
// ISA 7.12.2 f32 layouts:
//   A 16x4 : lane L -> M = L%16 ; VGPR0 holds K = (L<16?0:2), VGPR1 K = +1
//   B 4x16 : lane L -> N = L%16 ; VGPR0 holds K = (L<16?0:2), VGPR1 K = +1
//   C/D    : VGPR r -> M = r + (L<16?0:8), N = L%16
#define TM 2
#define TN 4
__global__ void __launch_bounds__(256, 1)
k_gemm_wmma(const float* __restrict__ A, const float* __restrict__ W,
            float* __restrict__ C, int M, int K, int Nn) {
  const int lane  = threadIdx.x & 31;
  const int wave  = threadIdx.x >> 5;
  const int waves = blockDim.x >> 5;
  const int Mtiles = M >> 4;                        // M is a multiple of 16
  const int nt0 = blockIdx.x * TN;                  // grid.x = Nn/(16*TN)
  const int mt0 = (blockIdx.y * waves + wave) * TM;
  if (mt0 >= Mtiles) return;                        // wave-uniform exit

  const int mn    = lane & 15;
  const int khalf = (lane >> 4) << 1;               // 0 or 2

  // base A pointer for tile 0; other tiles use scalar-uniform row offsets,
  // clamped for the (wave-uniform) partial last block
  const float* Abase = A + (size_t)(mt0 * 16 + mn) * K;
  size_t aoff[TM];
#pragma unroll
  for (int i = 0; i < TM; ++i) {
    int mt = mt0 + i; if (mt >= Mtiles) mt = Mtiles - 1;
    aoff[i] = (size_t)(mt - mt0) * 16 * (size_t)K;
  }
  const float* Wc = W + (size_t)(nt0 * 16 + mn);

  v8f acc[TM][TN];
#pragma unroll
  for (int i = 0; i < TM; ++i)
#pragma unroll
    for (int j = 0; j < TN; ++j) acc[i][j] = (v8f){};

#pragma unroll 1
  for (int k0 = 0; k0 < K; k0 += 4) {
    v2f a[TM], b[TN];
#pragma unroll
    for (int i = 0; i < TM; ++i) {
      a[i].x = Abase[aoff[i] + k0 + khalf + 0];
      a[i].y = Abase[aoff[i] + k0 + khalf + 1];
    }
#pragma unroll
    for (int j = 0; j < TN; ++j) {
      b[j].x = Wc[(size_t)(k0 + khalf + 0) * Nn + j * 16];
      b[j].y = Wc[(size_t)(k0 + khalf + 1) * Nn + j * 16];
    }
#pragma unroll
    for (int i = 0; i < TM; ++i)
#pragma unroll
      for (int j = 0; j < TN; ++j)
        acc[i][j] = __builtin_amdgcn_wmma_f32_16x16x4_f32(
            false, a[i], false, b[j], (short)0, acc[i][j], false, false);
  }

  const int rowoff = (lane >> 4) << 3;              // +8 for upper half-lanes
#pragma unroll
  for (int i = 0; i < TM; ++i) {
    if (mt0 + i >= Mtiles) break;                   // wave-uniform
    float* Cr = C + (size_t)((mt0 + i) * 16 + rowoff) * Nn + (size_t)(nt0 * 16 + mn);
#pragma unroll
    for (int j = 0; j < TN; ++j)
#pragma unroll
      for (int r = 0; r < 8; ++r)
        Cr[(size_t)r * Nn + j * 16] = acc[i][j][r];
  }
}

// ---------------- per-edge attention scores (wave per edge, float4) --------
__global__ void k_edge_scores(const float* __restrict__ xl, const float* __restrict__ xr,
                              const int* __restrict__ ei, const float* __restrict__ eattr,
                              const float* __restrict__ eaSum,
                              const float* __restrict__ We, const float* __restrict__ att,
                              float* __restrict__ score, float* __restrict__ smax,
                              int E, int N, int H, int Cshift) {
  const int lane = threadIdx.x & 31;
  const int wave = threadIdx.x >> 5;
  const int e = blockIdx.x * (blockDim.x >> 5) + wave;
  const int E2 = E + N;
  if (e >= E2) return;

  int src, dst; float4 ea;
  if (e < E) {
    src = ei[e]; dst = ei[E + e];
    ea = ((const float4*)eattr)[e];
  } else {                                  // self loop, edge_attr = column mean
    src = dst = e - E;
    const float inv = 1.0f / (float)E;
    ea = make_float4(eaSum[0] * inv, eaSum[1] * inv, eaSum[2] * inv, eaSum[3] * inv);
  }

  const int HC = H << Cshift;
  const int G  = HC >> 2;                   // float4 groups per row
  const float4* xls = (const float4*)xl + (size_t)src * G;
  const float4* xrd = (const float4*)xr + (size_t)dst * G;
  const float4* We4 = (const float4*)We;
  const float4* at4 = (const float4*)att;

  float acc[4] = {0.f, 0.f, 0.f, 0.f};
  for (int q = lane; q < G; q += 32) {
    const float4 l = xls[q], r = xrd[q];
    const float4 w0 = We4[q], w1 = We4[G + q], w2 = We4[2 * G + q], w3 = We4[3 * G + q];
    const float4 at = at4[q];
    const int h = (q << 2) >> Cshift;       // same head for all 4 components
    float m, lr, part = 0.f;
    m = l.x + r.x + ea.x * w0.x + ea.y * w1.x + ea.z * w2.x + ea.w * w3.x;
    lr = (m > 0.f) ? m : 0.2f * m; part += lr * at.x;
    m = l.y + r.y + ea.x * w0.y + ea.y * w1.y + ea.z * w2.y + ea.w * w3.y;
    lr = (m > 0.f) ? m : 0.2f * m; part += lr * at.y;
    m = l.z + r.z + ea.x * w0.z + ea.y * w1.z + ea.z * w2.z + ea.w * w3.z;
    lr = (m > 0.f) ? m : 0.2f * m; part += lr * at.z;
    m = l.w + r.w + ea.x * w0.w + ea.y * w1.w + ea.z * w2.w + ea.w * w3.w;
    lr = (m > 0.f) ? m : 0.2f * m; part += lr * at.w;
    acc[h] += part;
  }
  for (int h = 0; h < H; ++h) {
    float v = waveSum(acc[h]);
    if (lane == 0) {
      score[(size_t)e * H + h] = v;
      atomicMaxF(&smax[(size_t)dst * H + h], v);
    }
  }
}

// ---------------- exp(score - max) and denominator ----------------
__global__ void k_edge_exp(const int* __restrict__ ei, float* __restrict__ sc_ex,
                           const float* __restrict__ smax, float* __restrict__ denom,
                           int E, int N, int H) {
  long long idx = (long long)blockIdx.x * blockDim.x + threadIdx.x;
  const long long tot = (long long)(E + N) * H;
  if (idx >= tot) return;
  const long long e = idx / H;
  const int h = (int)(idx - e * H);
  const int dst = (e < E) ? ei[E + e] : (int)(e - E);
  const float ex = expf(sc_ex[idx] - smax[(size_t)dst * H + h]);
  sc_ex[idx] = ex;
  atomicAddF(&denom[(size_t)dst * H + h], ex);
}

// ---------------- weighted aggregation (wave per edge, float4 loads) -------
__global__ void k_edge_agg(const float* __restrict__ xl, const int* __restrict__ ei,
                           const float* __restrict__ ex, const float* __restrict__ denom,
                           float* __restrict__ out, int E, int N, int H, int Cshift) {
  const int lane = threadIdx.x & 31;
  const int wave = threadIdx.x >> 5;
  const int e = blockIdx.x * (blockDim.x >> 5) + wave;
  const int E2 = E + N;
  if (e >= E2) return;

  int src, dst;
  if (e < E) { src = ei[e]; dst = ei[E + e]; }
  else       { src = dst = e - E; }

  float alpha[4];
  for (int h = 0; h < H; ++h)
    alpha[h] = ex[(size_t)e * H + h] / (denom[(size_t)dst * H + h] + 1e-16f);

  const int HC = H << Cshift;
  const int G  = HC >> 2;
  const float4* xls = (const float4*)xl + (size_t)src * G;
  float* od = out + (size_t)dst * HC;
  for (int q = lane; q < G; q += 32) {
    const float4 v = xls[q];
    const float a = alpha[(q << 2) >> Cshift];
    float* p = od + (q << 2);
    atomicAddF(p + 0, a * v.x);
    atomicAddF(p + 1, a * v.y);
    atomicAddF(p + 2, a * v.z);
    atomicAddF(p + 3, a * v.w);
  }
}

// ---------------- LayerNorm (+ELU or +residual), wave per row, float4 ------
__global__ void k_layernorm(const float* __restrict__ in, const float* __restrict__ g,
                            const float* __restrict__ b, const float* __restrict__ res,
                            float* __restrict__ out, int rows, int F, int doElu) {
  const int lane = threadIdx.x & 31;
  const int wave = threadIdx.x >> 5;
  const int row = blockIdx.x * (blockDim.x >> 5) + wave;
  if (row >= rows) return;
  const int G = F >> 2;
  const float4* v4 = (const float4*)in + (size_t)row * G;
  float s = 0.f, ss = 0.f;
  for (int q = lane; q < G; q += 32) {
    float4 v = v4[q];
    s  += v.x + v.y + v.z + v.w;
    ss += v.x * v.x + v.y * v.y + v.z * v.z + v.w * v.w;
  }
  s = waveSum(s); ss = waveSum(ss);
  const float mean = s / (float)F;
  const float var  = ss / (float)F - mean * mean;      // biased, as jnp.var
  const float rstd = rsqrtf(var + 1e-5f);
  const float4* g4 = (const float4*)g;
  const float4* b4 = (const float4*)b;
  const float4* r4 = res ? (const float4*)res + (size_t)row * G : nullptr;
  float4* o4 = (float4*)out + (size_t)row * G;
  for (int q = lane; q < G; q += 32) {
    float4 v = v4[q], gg = g4[q], bb = b4[q], y;
    y.x = (v.x - mean) * rstd * gg.x + bb.x;
    y.y = (v.y - mean) * rstd * gg.y + bb.y;
    y.z = (v.z - mean) * rstd * gg.z + bb.z;
    y.w = (v.w - mean) * rstd * gg.w + bb.w;
    if (doElu) {
      y.x = (y.x > 0.f) ? y.x : (expf(y.x) - 1.f);
      y.y = (y.y > 0.f) ? y.y : (expf(y.y) - 1.f);
      y.z = (y.z > 0.f) ? y.z : (expf(y.z) - 1.f);
      y.w = (y.w > 0.f) ? y.w : (expf(y.w) - 1.f);
    }
    if (r4) {
      float4 r = r4[q];
      y.x += r.x; y.y += r.y; y.z += r.z; y.w += r.w;
    }
    o4[q] = y;
  }
}

// ---------------------------------------------------------------------------
extern "C" void kernel_launch(void* const* d_in, const int* in_sizes, int n_in,
                              void* d_out, int out_size, void* d_ws, size_t ws_size,
                              hipStream_t stream) {
  const float* x     = (const float*)d_in[0];
  const int*   ei    = (const int*)d_in[1];
  const float* eattr = (const float*)d_in[2];
  const float* Wl1   = (const float*)d_in[3];
  const float* Wr1   = (const float*)d_in[4];
  const float* We1   = (const float*)d_in[5];
  const float* att1  = (const float*)d_in[6];
  const float* b1    = (const float*)d_in[7];
  const float* g1    = (const float*)d_in[8];
  const float* be1   = (const float*)d_in[9];
  const float* Wl2   = (const float*)d_in[10];
  const float* Wr2   = (const float*)d_in[11];
  const float* We2   = (const float*)d_in[12];
  const float* att2  = (const float*)d_in[13];
  const float* b2    = (const float*)d_in[14];
  const float* g2    = (const float*)d_in[15];
  const float* be2   = (const float*)d_in[16];

  const int D = 256, HC = 512, F2 = 256;
  const int N  = in_sizes[0] / D;           // 10000 (multiple of 16)
  const int E  = in_sizes[2] / 4;           // 320000
  const int E2 = E + N;

  // workspace layout (floats)
  float* ws    = (float*)d_ws;
  float* xl1   = ws;                              // N*512
  float* xr1   = xl1 + (size_t)N * HC;            // N*512 (reused as out1)
  float* h1    = xr1 + (size_t)N * HC;            // N*512
  float* xl2   = h1 + (size_t)N * HC;             // N*256
  float* xr2   = xl2 + (size_t)N * F2;            // N*256 (reused as out2)
  float* exb   = xr2 + (size_t)N * F2;            // E2*4 (scores -> exp)
  float* smax  = exb + (size_t)E2 * 4;            // N*4
  float* denom = smax + (size_t)N * 4;            // N*4
  float* eaSum = denom + (size_t)N * 4;           // 4
  float* out1  = xr1;
  float* out2  = xr2;

  const float NEG_INF = -__builtin_inff();
  const int BLK = 256;                            // 8 waves
  const int WPB = BLK / 32;
  const int edgeBlocks = (E2 + WPB - 1) / WPB;
  const int Mtiles = N / 16;
  const int mblocks = (Mtiles + WPB * TM - 1) / (WPB * TM);

  // edge_attr column mean accumulator
  k_fill<<<1, 64, 0, stream>>>(eaSum, 0.f, 4);
  k_ea_accum<<<512, BLK, 0, stream>>>(eattr, eaSum, E);

  // ---- layer 1 ----
  {
    dim3 g1d(HC / (16 * TN), mblocks);
    k_gemm_wmma<<<g1d, BLK, 0, stream>>>(x, Wl1, xl1, N, D, HC);
    k_gemm_wmma<<<g1d, BLK, 0, stream>>>(x, Wr1, xr1, N, D, HC);
  }
  k_fill<<<(N * 4 + BLK - 1) / BLK, BLK, 0, stream>>>(smax, NEG_INF, (long long)N * 4);
  k_fill<<<(N * 4 + BLK - 1) / BLK, BLK, 0, stream>>>(denom, 0.f, (long long)N * 4);
  k_edge_scores<<<edgeBlocks, BLK, 0, stream>>>(xl1, xr1, ei, eattr, eaSum, We1, att1,
                                                exb, smax, E, N, 4, 7);
  {
    long long tot = (long long)E2 * 4;
    k_edge_exp<<<(int)((tot + BLK - 1) / BLK), BLK, 0, stream>>>(ei, exb, smax, denom,
                                                                E, N, 4);
  }
  k_bias_init<<<((long long)N * HC + BLK - 1) / BLK, BLK, 0, stream>>>(out1, b1,
                                                                      (long long)N * HC, HC);
  k_edge_agg<<<edgeBlocks, BLK, 0, stream>>>(xl1, ei, exb, denom, out1, E, N, 4, 7);
  k_layernorm<<<(N + WPB - 1) / WPB, BLK, 0, stream>>>(out1, g1, be1, nullptr, h1,
                                                       N, HC, /*elu=*/1);

  // ---- layer 2 (H=1, concat=False -> identity mean) ----
  {
    dim3 g2d(F2 / (16 * TN), mblocks);
    k_gemm_wmma<<<g2d, BLK, 0, stream>>>(h1, Wl2, xl2, N, HC, F2);
    k_gemm_wmma<<<g2d, BLK, 0, stream>>>(h1, Wr2, xr2, N, HC, F2);
  }
  k_fill<<<(N + BLK - 1) / BLK, BLK, 0, stream>>>(smax, NEG_INF, (long long)N);
  k_fill<<<(N + BLK - 1) / BLK, BLK, 0, stream>>>(denom, 0.f, (long long)N);
  k_edge_scores<<<edgeBlocks, BLK, 0, stream>>>(xl2, xr2, ei, eattr, eaSum, We2, att2,
                                                exb, smax, E, N, 1, 8);
  k_edge_exp<<<(E2 + BLK - 1) / BLK, BLK, 0, stream>>>(ei, exb, smax, denom, E, N, 1);
  k_bias_init<<<((long long)N * F2 + BLK - 1) / BLK, BLK, 0, stream>>>(out2, b2,
                                                                      (long long)N * F2, F2);
  k_edge_agg<<<edgeBlocks, BLK, 0, stream>>>(xl2, ei, exb, denom, out2, E, N, 1, 8);
  k_layernorm<<<(N + WPB - 1) / WPB, BLK, 0, stream>>>(out2, g2, be2, /*res=*/x,
                                                       (float*)d_out, N, F2, /*elu=*/0);
}